// SpectralConv2d_90443421319203
// MI455X (gfx1250) — compile-verified
//
#include <hip/hip_runtime.h>
#include <hip/hip_bf16.h>

typedef __attribute__((ext_vector_type(2))) float v2f;
typedef __attribute__((ext_vector_type(4))) float v4f;
typedef __attribute__((ext_vector_type(8))) float v8f;

#define LDSP 65
#define PI_F 3.14159265358979323846f

// D = A(16x4) * B(4x16) + C, fp32 WMMA (CDNA5)
__device__ __forceinline__ v8f wmma4(v2f a, v2f b, v8f c) {
  return __builtin_amdgcn_wmma_f32_16x16x4_f32(false, a, false, b, (short)0, c,
                                               false, false);
}

// Workspace layout (float offsets)
#define PLANE    4194304             // 16*64*64*64
#define OFF_FRE  0                   // DFT twiddle real  (64x64)
#define OFF_FIM  4096                // DFT twiddle imag
#define OFF_XRE  (8192)              // x_ft real  [b][i][u][v]
#define OFF_XIM  (8192 + 1*PLANE)    // x_ft imag
#define OFF_ARE  (8192 + 2*PLANE)    // out_ft real [x*64+y][b][o]
#define OFF_AIM  (8192 + 3*PLANE)    // out_ft imag
#define OFF_TRE  (8192 + 4*PLANE)    // out_ft real [b][o][x][y]
#define OFF_TIM  (8192 + 5*PLANE)    // out_ft imag

// ---------------------------------------------------------------- twiddles
__global__ void k_twiddle(float* __restrict__ ws) {
  int id = blockIdx.x * 256 + threadIdx.x;       // 4096
  int j = id >> 6, k = id & 63;
  int tt = (j * k) & 63;                         // e^{-2pi i jk/64} periodic
  float ang = -2.0f * PI_F * (float)tt / 64.0f;
  float s, c;
  sincosf(ang, &s, &c);
  ws[OFF_FRE + id] = c;
  ws[OFF_FIM + id] = s;
}

// ---------------------------------------------------------------- zero mid band
__global__ void k_zero(float* __restrict__ out) {
  size_t idx = (size_t)blockIdx.x * 256 + threadIdx.x;   // float4 units
  size_t b = idx / 393216, rem = idx - b * 393216;       // 384*4096/4 per batch
  v4f z = {0.f, 0.f, 0.f, 0.f};
  v4f* p = (v4f*)out;
  __builtin_nontemporal_store(z, &p[b * 524288 + 65536 + rem]);
}

// ---------------------------------------------------------------- forward FFT2
// One workgroup per (b,i) slice:  Y = F * X * F   (X real 64x64)
__global__ __launch_bounds__(256) void k_fft2(const float* __restrict__ x,
                                              float* __restrict__ ws, int s1base) {
  __shared__ float P0[64 * LDSP];
  __shared__ float P1[64 * LDSP];
  const float* Fre = ws + OFF_FRE;
  const float* Fim = ws + OFF_FIM;
  int b = blockIdx.x >> 6, i = blockIdx.x & 63;
  const float* X = x + (((size_t)b * 512 + s1base + i) << 12);
  int t = threadIdx.x;
  for (int j = t; j < 4096; j += 256) {
    int r = j >> 6, c = j & 63;
    P0[r * LDSP + c] = __builtin_nontemporal_load(&X[j]);
  }
  __syncthreads();
  int wave = t >> 5, ln = t & 15, hi = (t >> 4) & 1;
  // ---- stage 1: T = F * X   (Tre = Fre*X, Tim = Fim*X), 32 tile-jobs
  v8f acc[4];
#pragma unroll
  for (int jj = 0; jj < 4; ++jj) {
    int job = wave + (jj << 3);
    int plane = job >> 4, tile = job & 15;
    int tu = tile >> 2, tv = tile & 3;
    const float* Fg = plane ? Fim : Fre;
    v8f c = {};
    for (int k0 = 0; k0 < 64; k0 += 4) {
      int k = k0 + 2 * hi;
      v2f a;  a.x = Fg[(tu * 16 + ln) * 64 + k];
              a.y = Fg[(tu * 16 + ln) * 64 + k + 1];
      v2f bb; bb.x = P0[k * LDSP + tv * 16 + ln];
              bb.y = P0[(k + 1) * LDSP + tv * 16 + ln];
      c = wmma4(a, bb, c);
    }
    acc[jj] = c;
  }
  __syncthreads();
#pragma unroll
  for (int jj = 0; jj < 4; ++jj) {
    int job = wave + (jj << 3);
    int plane = job >> 4, tile = job & 15;
    int tu = tile >> 2, tv = tile & 3;
    float* P = plane ? P1 : P0;
#pragma unroll
    for (int r = 0; r < 8; ++r)
      P[(tu * 16 + r + 8 * hi) * LDSP + tv * 16 + ln] = acc[jj][r];
  }
  __syncthreads();
  // ---- stage 2: Y = T * F
  float* xre = ws + OFF_XRE + ((size_t)blockIdx.x << 12);
  float* xim = ws + OFF_XIM + ((size_t)blockIdx.x << 12);
#pragma unroll
  for (int jj = 0; jj < 4; ++jj) {
    int job = wave + (jj << 3);
    int plane = job >> 4, tile = job & 15;     // 0: Yre, 1: Yim
    int tu = tile >> 2, tv = tile & 3;
    v8f c = {};
    for (int k0 = 0; k0 < 64; k0 += 4) {
      int k = k0 + 2 * hi;
      v2f aR; aR.x = P0[(tu * 16 + ln) * LDSP + k];
              aR.y = P0[(tu * 16 + ln) * LDSP + k + 1];
      v2f aI; aI.x = P1[(tu * 16 + ln) * LDSP + k];
              aI.y = P1[(tu * 16 + ln) * LDSP + k + 1];
      v2f bR; bR.x = Fre[k * 64 + tv * 16 + ln];
              bR.y = Fre[(k + 1) * 64 + tv * 16 + ln];
      v2f bI; bI.x = Fim[k * 64 + tv * 16 + ln];
              bI.y = Fim[(k + 1) * 64 + tv * 16 + ln];
      if (plane == 0) {               // Yre = Tre*Fre - Tim*Fim
        c = wmma4(aR, bR, c);
        c = wmma4(aI, -bI, c);
      } else {                        // Yim = Tre*Fim + Tim*Fre
        c = wmma4(aR, bI, c);
        c = wmma4(aI, bR, c);
      }
    }
    float* dst = plane ? xim : xre;
#pragma unroll
    for (int r = 0; r < 8; ++r)
      dst[(tu * 16 + r + 8 * hi) * 64 + tv * 16 + ln] = c[r];
  }
}

// ---------------------------------------------------------------- einsum
// out[b,o,x,y] = sum_i xft[b,i,x,y] * w[i,o,x,y]  (w real -> 2 real GEMMs)
// Block: (x, y-block of 16). 16 waves: wave = otile(4) x ygroup(4); 4 y each.
__global__ __launch_bounds__(512) void k_einsum(const float* __restrict__ w,
                                                float* __restrict__ ws) {
  __shared__ float Wl[4 * 64 * 17];     // [k][o][ypad]
  __shared__ float Al[2 * 4 * 16 * 17]; // [plane][k][b][ypad]
  const float* xre = ws + OFF_XRE;
  const float* xim = ws + OFF_XIM;
  float* are = ws + OFF_ARE;
  float* aim = ws + OFF_AIM;
  int x = blockIdx.x, yb = blockIdx.y;
  int t = threadIdx.x;
  int wave = t >> 5, ln = t & 15, hi = (t >> 4) & 1;
  int ot = wave & 3, yg = wave >> 2;
  v8f zero = {};
  v8f cre[4] = {zero, zero, zero, zero};
  v8f cim[4] = {zero, zero, zero, zero};
  for (int ic = 0; ic < 16; ++ic) {
    for (int j = t; j < 4096; j += 512) {               // weights: 4k x 64o x 16y
      int c0 = j & 15, row = j >> 4;
      int k = row >> 6, o = row & 63;
      size_t g = ((((size_t)(ic * 4 + k) * 64 + o) * 64 + x) << 6) + yb * 16 + c0;
      Wl[row * 17 + c0] = __builtin_nontemporal_load(&w[g]);
    }
    for (int j = t; j < 2048; j += 512) {               // xft: 2 planes x 4k x 16b x 16y
      int c0 = j & 15, row = j >> 4;
      int plane = row >> 6, rr = row & 63;
      int k = rr >> 4, bb = rr & 15;
      const float* src = plane ? xim : xre;
      Al[row * 17 + c0] =
          src[((((size_t)bb * 64 + ic * 4 + k) * 64 + x) << 6) + yb * 16 + c0];
    }
    __syncthreads();
    int k2 = 2 * hi;
#pragma unroll
    for (int yy = 0; yy < 4; ++yy) {
      int yl = yg * 4 + yy;
      v2f bf;
      bf.x = Wl[(k2 * 64 + ot * 16 + ln) * 17 + yl];
      bf.y = Wl[((k2 + 1) * 64 + ot * 16 + ln) * 17 + yl];
      v2f aR;
      aR.x = Al[(k2 * 16 + ln) * 17 + yl];
      aR.y = Al[((k2 + 1) * 16 + ln) * 17 + yl];
      v2f aI;
      aI.x = Al[(64 + k2 * 16 + ln) * 17 + yl];
      aI.y = Al[(64 + (k2 + 1) * 16 + ln) * 17 + yl];
      cre[yy] = wmma4(aR, bf, cre[yy]);
      cim[yy] = wmma4(aI, bf, cim[yy]);
    }
    __syncthreads();
  }
#pragma unroll
  for (int yy = 0; yy < 4; ++yy) {
    int y = yb * 16 + yg * 4 + yy;
    size_t base = ((size_t)(x * 64 + y)) << 10;   // [xy][b][o]
#pragma unroll
    for (int r = 0; r < 8; ++r) {
      int m = r + 8 * hi;
      are[base + m * 64 + ot * 16 + ln] = cre[yy][r];
      aim[base + m * 64 + ot * 16 + ln] = cim[yy][r];
    }
  }
}

// ---------------------------------------------------------------- transpose
// [x*64+y][b][o]  ->  [b][o][x][y]   (LDS 64x64 tile, both planes)
__global__ __launch_bounds__(256) void k_transpose(float* __restrict__ ws) {
  __shared__ float Tl[64 * LDSP];
  int b = blockIdx.x >> 6, x = blockIdx.x & 63;
  int t = threadIdx.x;
  for (int p = 0; p < 2; ++p) {
    const float* src = ws + (p ? OFF_AIM : OFF_ARE);
    float* dst = ws + (p ? OFF_TIM : OFF_TRE);
    for (int j = t; j < 4096; j += 256) {
      int y = j >> 6, o = j & 63;
      Tl[y * LDSP + o] = src[((size_t)(x * 64 + y) << 10) + b * 64 + o];
    }
    __syncthreads();
    for (int j = t; j < 4096; j += 256) {
      int o = j >> 6, y = j & 63;
      dst[((size_t)((b * 64 + o) * 64 + x) << 6) + y] = Tl[y * LDSP + o];
    }
    __syncthreads();
  }
}

// ---------------------------------------------------------------- inverse FFT2
// out[b, base+o, :, :] = real(conj(F) * Z * conj(F)) / 4096
__global__ __launch_bounds__(256) void k_ifft2(float* __restrict__ ws,
                                               float* __restrict__ out, int s1base) {
  __shared__ float P0[64 * LDSP];
  __shared__ float P1[64 * LDSP];
  const float* Fre = ws + OFF_FRE;
  const float* Fim = ws + OFF_FIM;
  int b = blockIdx.x >> 6, o = blockIdx.x & 63;
  const float* zre = ws + OFF_TRE + ((size_t)blockIdx.x << 12);
  const float* zim = ws + OFF_TIM + ((size_t)blockIdx.x << 12);
  int t = threadIdx.x;
  for (int j = t; j < 4096; j += 256) {
    int r = j >> 6, c = j & 63;
    P0[r * LDSP + c] = zre[j];
    P1[r * LDSP + c] = zim[j];
  }
  __syncthreads();
  int wave = t >> 5, ln = t & 15, hi = (t >> 4) & 1;
  // stage 1: P = conj(F) * Z
  v8f acc[4];
#pragma unroll
  for (int jj = 0; jj < 4; ++jj) {
    int job = wave + (jj << 3);
    int plane = job >> 4, tile = job & 15;
    int tu = tile >> 2, tv = tile & 3;
    v8f c = {};
    for (int k0 = 0; k0 < 64; k0 += 4) {
      int k = k0 + 2 * hi;
      v2f aR; aR.x = Fre[(tu * 16 + ln) * 64 + k];
              aR.y = Fre[(tu * 16 + ln) * 64 + k + 1];
      v2f aI; aI.x = Fim[(tu * 16 + ln) * 64 + k];
              aI.y = Fim[(tu * 16 + ln) * 64 + k + 1];
      v2f bR; bR.x = P0[k * LDSP + tv * 16 + ln];
              bR.y = P0[(k + 1) * LDSP + tv * 16 + ln];
      v2f bI; bI.x = P1[k * LDSP + tv * 16 + ln];
              bI.y = P1[(k + 1) * LDSP + tv * 16 + ln];
      if (plane == 0) {               // Pre = Fre*Zre + Fim*Zim
        c = wmma4(aR, bR, c);
        c = wmma4(aI, bI, c);
      } else {                        // Pim = Fre*Zim - Fim*Zre
        c = wmma4(aR, bI, c);
        c = wmma4(aI, -bR, c);
      }
    }
    acc[jj] = c;
  }
  __syncthreads();
#pragma unroll
  for (int jj = 0; jj < 4; ++jj) {
    int job = wave + (jj << 3);
    int plane = job >> 4, tile = job & 15;
    int tu = tile >> 2, tv = tile & 3;
    float* P = plane ? P1 : P0;
#pragma unroll
    for (int r = 0; r < 8; ++r)
      P[(tu * 16 + r + 8 * hi) * LDSP + tv * 16 + ln] = acc[jj][r];
  }
  __syncthreads();
  // stage 2: Qre = Pre*Fre + Pim*Fim, scaled
  float* dst = out + (((size_t)b * 512 + s1base + o) << 12);
#pragma unroll
  for (int jj = 0; jj < 2; ++jj) {
    int tile = wave + (jj << 3);
    int tu = tile >> 2, tv = tile & 3;
    v8f c = {};
    for (int k0 = 0; k0 < 64; k0 += 4) {
      int k = k0 + 2 * hi;
      v2f aR; aR.x = P0[(tu * 16 + ln) * LDSP + k];
              aR.y = P0[(tu * 16 + ln) * LDSP + k + 1];
      v2f aI; aI.x = P1[(tu * 16 + ln) * LDSP + k];
              aI.y = P1[(tu * 16 + ln) * LDSP + k + 1];
      v2f bR; bR.x = Fre[k * 64 + tv * 16 + ln];
              bR.y = Fre[(k + 1) * 64 + tv * 16 + ln];
      v2f bI; bI.x = Fim[k * 64 + tv * 16 + ln];
              bI.y = Fim[(k + 1) * 64 + tv * 16 + ln];
      c = wmma4(aR, bR, c);
      c = wmma4(aI, bI, c);
    }
    const float sc = 1.0f / 4096.0f;
#pragma unroll
    for (int r = 0; r < 8; ++r)
      __builtin_nontemporal_store(c[r] * sc,
          &dst[(tu * 16 + r + 8 * hi) * 64 + tv * 16 + ln]);
  }
}

extern "C" void kernel_launch(void* const* d_in, const int* in_sizes, int n_in,
                              void* d_out, int out_size, void* d_ws, size_t ws_size,
                              hipStream_t stream) {
  const float* x  = (const float*)d_in[0];
  const float* w1 = (const float*)d_in[1];
  const float* w2 = (const float*)d_in[2];
  float* out = (float*)d_out;
  float* ws  = (float*)d_ws;

  k_twiddle<<<16, 256, 0, stream>>>(ws);
  k_zero<<<24576, 256, 0, stream>>>(out);           // out[:, 64:448] = 0
  for (int h = 0; h < 2; ++h) {
    const float* w = h ? w2 : w1;
    int base = h ? 448 : 0;
    k_fft2<<<1024, 256, 0, stream>>>(x, ws, base);
    k_einsum<<<dim3(64, 4), 512, 0, stream>>>(w, ws);
    k_transpose<<<1024, 256, 0, stream>>>(ws);
    k_ifft2<<<1024, 256, 0, stream>>>(ws, out, base);
  }
}